// TransitionBuffer_63178968924201
// MI455X (gfx1250) — compile-verified
//
#include <hip/hip_runtime.h>
#include <cstdint>

typedef unsigned int u32;
typedef unsigned long long u64;
typedef u32 v4u __attribute__((ext_vector_type(4)));
typedef int  v8i __attribute__((ext_vector_type(8)));
typedef int  v4i __attribute__((ext_vector_type(4)));

#define NBUCKET  2048
#define CAND_CAP 8192
#define FEAT     256

// ---------------- threefry2x32 with JAX key(42) = (0, 42) ----------------
__device__ __forceinline__ u32 rotl32(u32 v, u32 n) { return (v << n) | (v >> (32u - n)); }

__device__ __forceinline__ void threefry2x32_k42(u32 x0, u32 x1, u32& o0, u32& o1) {
    const u32 k0 = 0u, k1 = 42u;
    const u32 k2 = k0 ^ k1 ^ 0x1BD11BDAu;
    x0 += k0; x1 += k1;
#define TF_R4(a, b, c, d)                                   \
    { x0 += x1; x1 = rotl32(x1, a); x1 ^= x0;               \
      x0 += x1; x1 = rotl32(x1, b); x1 ^= x0;               \
      x0 += x1; x1 = rotl32(x1, c); x1 ^= x0;               \
      x0 += x1; x1 = rotl32(x1, d); x1 ^= x0; }
    TF_R4(13, 15, 26, 6);  x0 += k1; x1 += k2 + 1u;
    TF_R4(17, 29, 16, 24); x0 += k2; x1 += k0 + 2u;
    TF_R4(13, 15, 26, 6);  x0 += k0; x1 += k1 + 3u;
    TF_R4(17, 29, 16, 24); x0 += k1; x1 += k2 + 4u;
    TF_R4(13, 15, 26, 6);  x0 += k2; x1 += k0 + 5u;
#undef TF_R4
    o0 = x0; o1 = x1;
}

__device__ __forceinline__ u32 new_size_of(const int* p_size, int max_size, int data_len) {
    long long ns = (long long)(*p_size) + (long long)data_len;
    long long ms = (long long)max_size;
    return (u32)(ns < ms ? ns : ms);
}

// ---------------- pass 0: zero histogram + selection/counter words ----------------
__global__ void k_init(u32* head) {
    int i = blockIdx.x * blockDim.x + threadIdx.x;
    if (i < NBUCKET + 8) head[i] = 0u;
}

// ---------------- pass 1: threefry + LDS histogram of top-11 mantissa bits ----------------
__global__ void k_rng_hist(u32* __restrict__ hist, const int* __restrict__ p_size,
                           int max_size, int data_len) {
    __shared__ u32 lh[NBUCKET];
    for (int i = threadIdx.x; i < NBUCKET; i += blockDim.x) lh[i] = 0u;
    __syncthreads();

    const u32 new_size = new_size_of(p_size, max_size, data_len);
    const int half = max_size >> 1;  // JAX splits iota(N) into two counter halves
    for (int p = blockIdx.x * blockDim.x + threadIdx.x; p < half;
         p += gridDim.x * blockDim.x) {
        u32 o0, o1;
        threefry2x32_k42((u32)p, (u32)(p + half), o0, o1);
        // uniform value = bitcast(0x3f800000 | (bits>>9)) - 1  -> order == mantissa order
        if ((u32)p < new_size)          atomicAdd(&lh[o0 >> 21], 1u);
        if ((u32)(p + half) < new_size) atomicAdd(&lh[o1 >> 21], 1u);
    }
    __syncthreads();
    for (int i = threadIdx.x; i < NBUCKET; i += blockDim.x)
        if (lh[i]) atomicAdd(&hist[i], lh[i]);
}

// ---------------- pass 2: find threshold bucket (suffix scan from the top) ----------------
__global__ void k_select(const u32* __restrict__ hist, u32* __restrict__ sel, int K) {
    if (blockIdx.x == 0 && threadIdx.x == 0) {
        u32 cum = 0;
        int b;
        for (b = NBUCKET - 1; b >= 0; --b) {
            cum += hist[b];
            if ((int)cum >= K) break;
        }
        sel[0] = (b < 0) ? 0u : (u32)b;  // threshold bucket (inclusive)
    }
}

// ---------------- pass 3: compact candidates (mantissa, index) ----------------
__global__ void k_compact(u32* __restrict__ sel, u32* __restrict__ cand,
                          const int* __restrict__ p_size, int max_size, int data_len) {
    const u32 B = sel[0];
    const u32 new_size = new_size_of(p_size, max_size, data_len);
    const int half = max_size >> 1;
    for (int p = blockIdx.x * blockDim.x + threadIdx.x; p < half;
         p += gridDim.x * blockDim.x) {
        u32 o0, o1;
        threefry2x32_k42((u32)p, (u32)(p + half), o0, o1);
        u32 m0 = o0 >> 9, m1 = o1 >> 9;
        if ((u32)p < new_size && (m0 >> 12) >= B) {
            u32 pos = atomicAdd(&sel[1], 1u);
            if (pos < CAND_CAP) { cand[2 * pos] = m0; cand[2 * pos + 1] = (u32)p; }
        }
        if ((u32)(p + half) < new_size && (m1 >> 12) >= B) {
            u32 pos = atomicAdd(&sel[1], 1u);
            if (pos < CAND_CAP) { cand[2 * pos] = m1; cand[2 * pos + 1] = (u32)(p + half); }
        }
    }
}

// ---------------- pass 4: bitonic sort 8192 packed keys in LDS, emit top-K indices ----------------
// key = (mantissa << 32) | ~index  -> descending key == descending value, ties by smaller index
__global__ void k_sort(const u32* __restrict__ cand, const u32* __restrict__ sel,
                       u32* __restrict__ topIdx, int K) {
    extern __shared__ u64 keys[];
    const int N = CAND_CAP;
    const int T = blockDim.x;
    u32 cnt = sel[1];
    if (cnt > (u32)N) cnt = (u32)N;

    for (int i = threadIdx.x; i < N; i += T)
        keys[i] = (i < (int)cnt)
                      ? (((u64)cand[2 * i] << 32) | (u32)(~cand[2 * i + 1]))
                      : 0ull;
    __syncthreads();

    for (u32 k = 2; k <= (u32)N; k <<= 1) {
        for (u32 j = k >> 1; j > 0; j >>= 1) {
            for (u32 i = threadIdx.x; i < (u32)N; i += T) {
                u32 ixj = i ^ j;
                if (ixj > i) {
                    u64 a = keys[i], b = keys[ixj];
                    bool blockAsc = ((i & k) == 0);
                    bool sw = blockAsc ? (a < b) : (a > b);  // overall descending
                    if (sw) { keys[i] = b; keys[ixj] = a; }
                }
            }
            __syncthreads();
        }
    }
    for (int i = threadIdx.x; i < K; i += T) topIdx[i] = ~((u32)keys[i]);
}

// ---------------- pass 5: TDM gather of 8 rows/block + window patch + TDM store ----------------
struct GatherSmem {
    float tile[8 * FEAT];  // first member -> assumed LDS offset 0 for the D# lds_addr
    u32 gi[8];
    u32 off[8];
};

__global__ void k_gather(const float* __restrict__ buf, const float* __restrict__ trans,
                         const u32* __restrict__ topIdx, float* __restrict__ out,
                         const int* __restrict__ p_ins, int max_size, int data_len) {
    __shared__ GatherSmem sm;
    const int base = blockIdx.x * 8;
    const u32 ins = (u32)(*p_ins);

    if (threadIdx.x < 8) {
        u32 gi = topIdx[base + threadIdx.x];
        u32 off = (gi >= ins) ? (gi - ins) : (gi + (u32)max_size - ins);
        sm.gi[threadIdx.x] = gi;
        sm.off[threadIdx.x] = (off < (u32)data_len) ? off : 0xFFFFFFFFu;
    }
    __syncthreads();

    if (threadIdx.x < 32) {  // wave 0 only: one TDM gather for 8 rows
        u32 idxs[8];
#pragma unroll
        for (int r = 0; r < 8; ++r)
            idxs[r] = (sm.off[r] == 0xFFFFFFFFu) ? sm.gi[r] : 0u;  // window rows: junk, patched later

        u64 ga = (u64)(uintptr_t)buf;
        v4u g0;
        g0.x = 0xC0000001u;                 // count=1, gather_index_size=32b, gather_mode=1
        g0.y = 0u;                          // lds_addr = 0 (sm.tile)
        g0.z = (u32)ga;                     // global_addr[31:0]
        g0.w = (u32)((ga >> 32) & 0x01FFFFFFull) | 0x80000000u;  // addr[56:32] | type=2
        v8i g1;
        g1[0] = 0x00020000;                                   // data_size = 4B
        g1[1] = (int)(((u32)FEAT & 0xFFFFu) << 16);           // tensor_dim0 = 256 (lo16)
        g1[2] = (int)(((u32)max_size & 0xFFFFu) << 16);       // tensor_dim1 lo16
        g1[3] = (int)((((u32)FEAT) << 16) |                   // tile_dim0 = 256
                      (((u32)max_size >> 16) & 0xFFFFu));     // tensor_dim1 hi16
        g1[4] = 8;                                            // tile_dim1 = #valid indices
        g1[5] = FEAT;                                         // tensor_dim0_stride = 256
        g1[6] = 0; g1[7] = 0;
        v4i g2, g3;
        g2[0] = (int)idxs[0]; g2[1] = (int)idxs[1]; g2[2] = (int)idxs[2]; g2[3] = (int)idxs[3];
        g3[0] = (int)idxs[4]; g3[1] = (int)idxs[5]; g3[2] = (int)idxs[6]; g3[3] = (int)idxs[7];
        v8i g4;
        g4[0] = 0; g4[1] = 0; g4[2] = 0; g4[3] = 0;
        g4[4] = 0; g4[5] = 0; g4[6] = 0; g4[7] = 0;
        __builtin_amdgcn_tensor_load_to_lds(g0, g1, g2, g3, g4, 0);
        __builtin_amdgcn_s_wait_tensorcnt(0);
    }
    __syncthreads();

    // patch rows that fall inside the ring-buffer insert window from `transition`
    for (int e = threadIdx.x; e < 8 * FEAT; e += blockDim.x) {
        int r = e >> 8;
        u32 off = sm.off[r];
        if (off != 0xFFFFFFFFu)
            sm.tile[e] = trans[(size_t)off * FEAT + (e & (FEAT - 1))];
    }
    __syncthreads();

    if (threadIdx.x < 32) {  // wave 0: DMA the 8x256 tile to the contiguous output
        u64 sa = (u64)(uintptr_t)(out + (size_t)base * FEAT);
        v4u h0;
        h0.x = 0x00000001u;  // count=1, no gather
        h0.y = 0u;           // lds_addr = 0
        h0.z = (u32)sa;
        h0.w = (u32)((sa >> 32) & 0x01FFFFFFull) | 0x80000000u;
        v8i h1;
        h1[0] = 0x00020000;                        // data_size = 4B
        h1[1] = (int)(((u32)FEAT) << 16);          // tensor_dim0 = 256
        h1[2] = (int)(8u << 16);                   // tensor_dim1 = 8
        h1[3] = (int)(((u32)FEAT) << 16);          // tile_dim0 = 256
        h1[4] = 8;                                 // tile_dim1 = 8
        h1[5] = FEAT;                              // tensor_dim0_stride = 256
        h1[6] = 0; h1[7] = 0;
        v4i z; z[0] = 0; z[1] = 0; z[2] = 0; z[3] = 0;
        v8i z8;
        z8[0] = 0; z8[1] = 0; z8[2] = 0; z8[3] = 0;
        z8[4] = 0; z8[5] = 0; z8[6] = 0; z8[7] = 0;
        __builtin_amdgcn_tensor_store_from_lds(h0, h1, z, z, z8, 0);
        __builtin_amdgcn_s_wait_tensorcnt(0);
    }
}

// ---------------- host launcher ----------------
extern "C" void kernel_launch(void* const* d_in, const int* in_sizes, int n_in,
                              void* d_out, int out_size, void* d_ws, size_t ws_size,
                              hipStream_t stream) {
    const float* buf   = (const float*)d_in[0];
    const float* trans = (const float*)d_in[1];
    const int* p_ins   = (const int*)d_in[2];
    const int* p_size  = (const int*)d_in[3];
    (void)n_in; (void)ws_size;

    const int max_size = in_sizes[0] / FEAT;   // 1,000,000
    const int data_len = in_sizes[1] / FEAT;   // 16,384
    const int K        = out_size / FEAT;      // 4,096

    // workspace layout
    u32* hist   = (u32*)d_ws;                                   // 2048 u32
    u32* sel    = hist + NBUCKET;                               // [0]=bucket, [1]=cand count
    u32* cand   = (u32*)((char*)d_ws + 8448);                   // 8192 x (m, idx)
    u32* topIdx = (u32*)((char*)d_ws + 8448 + CAND_CAP * 8);    // 4096 u32

    k_init<<<(NBUCKET + 8 + 255) / 256, 256, 0, stream>>>(hist);
    k_rng_hist<<<512, 256, 0, stream>>>(hist, p_size, max_size, data_len);
    k_select<<<1, 32, 0, stream>>>(hist, sel, K);
    k_compact<<<512, 256, 0, stream>>>(sel, cand, p_size, max_size, data_len);
    k_sort<<<1, 1024, CAND_CAP * sizeof(u64), stream>>>(cand, sel, topIdx, K);
    k_gather<<<K / 8, 256, 0, stream>>>(buf, trans, topIdx, (float*)d_out,
                                        p_ins, max_size, data_len);
}